// GNN_9869834846215
// MI455X (gfx1250) — compile-verified
//
#include <hip/hip_runtime.h>
#include <math.h>

typedef float v2f __attribute__((ext_vector_type(2)));
typedef float v8f __attribute__((ext_vector_type(8)));

#define FDIM 128   // F == H == 128 for this problem

// ---------------------------------------------------------------------------
// Degree / normalization kernels
// ---------------------------------------------------------------------------
__global__ void k_init_deg(float* __restrict__ deg, int N) {
    int i = blockIdx.x * blockDim.x + threadIdx.x;
    if (i < N) deg[i] = 1.0f;                 // self-loop contributes 1
}

__global__ void k_count_deg(const int* __restrict__ dst, float* __restrict__ deg, int E) {
    int stride = gridDim.x * blockDim.x;
    for (int e = blockIdx.x * blockDim.x + threadIdx.x; e < E; e += stride) {
        atomicAdd(&deg[dst[e]], 1.0f);
    }
}

__global__ void k_dinv(float* __restrict__ deg, int N) {
    int i = blockIdx.x * blockDim.x + threadIdx.x;
    if (i < N) {
        float d = deg[i];
        deg[i] = (d > 0.0f) ? rsqrtf(d) : 0.0f;   // in-place: deg -> dinv
    }
}

// ---------------------------------------------------------------------------
// Fused WMMA GEMM + self-loop epilogue:
//   h           = (RELU_A ? relu(A) : A) @ W          -> outH  (raw, for gathers)
//   agg_init    = dinv[row]^2 * h + bias[col]         -> outAgg (self-loop + bias folded)
//
// W is staged once per block into LDS, pre-swizzled into B-fragment order:
//   float2 entry (kstep, tile, lane) = { W[4k+2h][c], W[4k+2h+1][c] },
//   h = lane>>4, c = tile*16 + (lane&15). Inner loop: one conflict-free
//   ds_load_b64 per tile per K-step (32 lanes x 2 dwords = 64 banks once).
//
// Fragment layouts per CDNA5 ISA 7.12.2 (V_WMMA_F32_16X16X4_F32):
//   A (16x4):  lane<16 -> row=lane, K=k..k+1 ; lane>=16 -> row=lane-16, K=k+2..k+3
//   B (4x16):  lane<16 -> col=lane, K=k,k+1  ; lane>=16 -> col=lane-16, K=k+2,k+3
//   C/D(16x16): vgpr r: lanes 0-15 -> M=r, lanes 16-31 -> M=8+r; N = lane&15
// ---------------------------------------------------------------------------
template <bool RELU_A>
__global__ void k_gemm128_fused(const float* __restrict__ A,
                                const float* __restrict__ W,
                                const float* __restrict__ bias,
                                const float* __restrict__ dinv,
                                float* __restrict__ outH,
                                float* __restrict__ outAgg, int M) {
    __shared__ float sW[FDIM * FDIM];        // 64 KB, B-fragment-major (float2 entries)

    const int lane = threadIdx.x & 31;
    const int half = lane >> 4;              // 0: lanes 0-15, 1: lanes 16-31
    const int l16  = lane & 15;

    // ---- stage W into LDS in fragment order (all waves participate) ----
    // entries: kstep(32) x tile(8) x lane(32), float2 each
    for (int e = threadIdx.x; e < 32 * 8 * 32; e += blockDim.x) {
        int kstep = e >> 8;                   // /256
        int rem   = e & 255;
        int t     = rem >> 5;
        int ln    = rem & 31;
        int h     = ln >> 4;
        int c     = t * 16 + (ln & 15);
        int krow  = kstep * 4 + 2 * h;
        float2 v;
        v.x = W[(size_t)krow * FDIM + c];
        v.y = W[(size_t)(krow + 1) * FDIM + c];
        ((float2*)sW)[e] = v;
    }
    __syncthreads();

    const int waveId = (blockIdx.x * blockDim.x + threadIdx.x) >> 5;
    const int rowT   = waveId;               // 16-row tile index (wave-uniform)
    if (rowT * 16 >= M) return;              // wave-uniform: EXEC all-1 for WMMA

    const float* __restrict__ arow = A + (size_t)(rowT * 16 + l16) * FDIM;

    v8f acc[8];
#pragma unroll
    for (int t = 0; t < 8; ++t) acc[t] = (v8f){0, 0, 0, 0, 0, 0, 0, 0};

    for (int kstep = 0; kstep < 32; ++kstep) {
        v2f a = *(const v2f*)(arow + kstep * 4 + 2 * half);   // global b64
        if (RELU_A) {
            a.x = fmaxf(a.x, 0.0f);
            a.y = fmaxf(a.y, 0.0f);
        }
        const v2f* __restrict__ bfrag = (const v2f*)sW + kstep * 256 + lane;
#pragma unroll
        for (int t = 0; t < 8; ++t) {
            v2f b = bfrag[t * 32];                            // ds_load_b64, no conflicts
            acc[t] = __builtin_amdgcn_wmma_f32_16x16x4_f32(
                false, a, false, b, (short)0, acc[t], false, false);
        }
    }

    // ---- fused epilogue: raw h + self-loop/bias-folded aggregation init ----
    float dv2[8];
#pragma unroll
    for (int r = 0; r < 8; ++r) {
        float dv = dinv[rowT * 16 + r + half * 8];
        dv2[r] = dv * dv;
    }
#pragma unroll
    for (int t = 0; t < 8; ++t) {
        const int col = t * 16 + l16;
        const float bc = bias[col];
#pragma unroll
        for (int r = 0; r < 8; ++r) {
            const int row = rowT * 16 + r + half * 8;
            const float h = acc[t][r];
            outH[(size_t)row * FDIM + col]   = h;
            outAgg[(size_t)row * FDIM + col] = dv2[r] * h + bc;
        }
    }
}

// ---------------------------------------------------------------------------
// Edge scatter-aggregate: out[dst,:] += dinv[src]*dinv[dst] * h[src,:]
// One wave per edge; each lane handles 4 contiguous features (float4 gather
// from the 512B source row + 4 global f32 atomics; all L2-resident).
// ---------------------------------------------------------------------------
__global__ void k_edge_agg(const float* __restrict__ h,
                           const float* __restrict__ dinv,
                           const int* __restrict__ src,
                           const int* __restrict__ dst,
                           float* __restrict__ out, int E) {
    int gid   = blockIdx.x * blockDim.x + threadIdx.x;
    int wave  = gid >> 5;
    int lane  = gid & 31;
    int nwave = (gridDim.x * blockDim.x) >> 5;
    for (int e = wave; e < E; e += nwave) {
        int s = src[e];
        int d = dst[e];
        float nrm = dinv[s] * dinv[d];
        float4 v = ((const float4*)(h + (size_t)s * FDIM))[lane];
        float* od = out + (size_t)d * FDIM + 4 * lane;
        atomicAdd(od + 0, nrm * v.x);
        atomicAdd(od + 1, nrm * v.y);
        atomicAdd(od + 2, nrm * v.z);
        atomicAdd(od + 3, nrm * v.w);
    }
}

// ---------------------------------------------------------------------------
// Launch
// ---------------------------------------------------------------------------
extern "C" void kernel_launch(void* const* d_in, const int* in_sizes, int n_in,
                              void* d_out, int out_size, void* d_ws, size_t ws_size,
                              hipStream_t stream) {
    const float* x   = (const float*)d_in[0];
    const int*   ei  = (const int*)d_in[1];     // edge_index, shape (2, E)
    const float* W1  = (const float*)d_in[2];
    const float* b1  = (const float*)d_in[3];
    const float* W2  = (const float*)d_in[4];
    const float* b2  = (const float*)d_in[5];
    float*       out = (float*)d_out;

    const int N = in_sizes[0] / FDIM;
    const int E = in_sizes[1] / 2;
    const int* src = ei;
    const int* dst = ei + E;

    // workspace layout: dinv[N] | bufH[N*128] (h1 then h2) | bufA[N*128] (agg1)
    char*  ws   = (char*)d_ws;
    float* dinv = (float*)ws;
    size_t off  = ((size_t)N * sizeof(float) + 255) & ~(size_t)255;
    float* bufH = (float*)(ws + off);
    float* bufA = (float*)(ws + off + (size_t)N * FDIM * sizeof(float));

    const int TPB = 256;

    // 1) normalization (cached, reused by both layers)
    k_init_deg<<<(N + TPB - 1) / TPB, TPB, 0, stream>>>(dinv, N);
    k_count_deg<<<2048, TPB, 0, stream>>>(dst, dinv, E);
    k_dinv<<<(N + TPB - 1) / TPB, TPB, 0, stream>>>(dinv, N);

    const int tiles      = (N + 15) / 16;              // 16-row tiles
    const int gemmBlocks = (tiles + (TPB / 32) - 1) / (TPB / 32);

    // 2) layer 1: h1 = x @ W1 ; agg1 init = dinv^2*h1 + b1 ; edges scatter-add
    k_gemm128_fused<false><<<gemmBlocks, TPB, 0, stream>>>(x, W1, b1, dinv, bufH, bufA, N);
    k_edge_agg<<<4096, TPB, 0, stream>>>(bufH, dinv, src, dst, bufA, E);

    // 3) layer 2: h2 = relu(agg1) @ W2 ; out init = dinv^2*h2 + b2 ; edges into d_out
    k_gemm128_fused<true><<<gemmBlocks, TPB, 0, stream>>>(bufA, W2, b2, dinv, bufH, out, N);
    k_edge_agg<<<4096, TPB, 0, stream>>>(bufH, dinv, src, dst, out, E);
}